// Net_59871844106652
// MI455X (gfx1250) — compile-verified
//
#include <hip/hip_runtime.h>

typedef __attribute__((ext_vector_type(2))) float v2f;
typedef __attribute__((ext_vector_type(8))) float v8f;

// ---------------- elementwise helpers ----------------

__global__ void k_zero(float* __restrict__ p, long long n) {
    long long i = (long long)blockIdx.x * blockDim.x + threadIdx.x;
    if (i < n) p[i] = 0.0f;
}

// deg -> rsqrt(max(deg,1)) in place
__global__ void k_rsqrt_norm(float* __restrict__ p, int n) {
    int i = blockIdx.x * blockDim.x + threadIdx.x;
    if (i < n) {
        float v = p[i];
        p[i] = rsqrtf(fmaxf(v, 1.0f));
    }
}

// ---------------- degree accumulation ----------------

__global__ void k_degrees(const int* __restrict__ src, const int* __restrict__ dst,
                          float* __restrict__ deg_out, float* __restrict__ deg_in, int E) {
    int e = blockIdx.x * blockDim.x + threadIdx.x;
    if (e < E) {
        atomicAdd(&deg_out[src[e]], 1.0f);
        atomicAdd(&deg_in[dst[e]], 1.0f);
    }
}

// ---------------- fp32 WMMA GEMM: P = (diag(ns) * X) @ W ----------------
// One wave32 computes one 16x16 output tile, looping K in steps of 4 with
// v_wmma_f32_16x16x4_f32. K and NC are compile-time so all load/store
// addresses are base + immediate (clause-friendly); K-loop unrolled 8x.
//   A (16x4 f32):  lane 0-15 -> M=lane;  VGPR0 = K = 2*half, VGPR1 = K = 2*half+1
//   B (4x16 f32):  lane 0-15 -> N=lane;  VGPR0 = K = 2*half, VGPR1 = K = 2*half+1
//   C/D (16x16):   VGPR r -> M = r + 8*half, N = lane&15
template <int K, int NC>
__global__ void k_gemm_scaled(const float* __restrict__ X, const float* __restrict__ ns,
                              const float* __restrict__ W, float* __restrict__ P,
                              int nrows) {
    const int m0   = blockIdx.x * 16;
    const int n0   = blockIdx.y * 16;
    const int lane = threadIdx.x;      // 0..31
    const int half = lane >> 4;        // 0 or 1
    const int lid  = lane & 15;

    const int  arow_raw = m0 + lid;
    const int  arow     = arow_raw < nrows ? arow_raw : (nrows - 1);
    const float scale   = arow_raw < nrows ? ns[arow] : 0.0f;  // zero kills OOB rows

    // A: per-lane row pointer, element pair starts at K-offset 2*half
    const float* __restrict__ xa = X + (size_t)arow * K + 2 * half;
    // B: per-lane column pointer, row starts at K-offset 2*half
    const float* __restrict__ wb = W + (size_t)(2 * half) * NC + n0 + lid;

    v8f acc = {};
#pragma unroll 8
    for (int k = 0; k < K; k += 4) {
        v2f a, b;
        const float2 av = *(const float2*)(xa + k);   // K = k+2h, k+2h+1
        a.x = av.x * scale;
        a.y = av.y * scale;
        b.x = wb[(size_t)k * NC];                     // K = k+2h
        b.y = wb[(size_t)(k + 1) * NC];               // K = k+2h+1
        acc = __builtin_amdgcn_wmma_f32_16x16x4_f32(
            /*neg_a=*/false, a, /*neg_b=*/false, b,
            /*c_mod=*/(short)0, acc, /*reuse_a=*/false, /*reuse_b=*/false);
    }

    // Store: per-lane base pointer; the 8 rows are base + immediate r*NC.
    float* __restrict__ pp = P + (size_t)(m0 + 8 * half) * NC + n0 + lid;
    if (m0 + 16 <= nrows) {            // full tile (always true when nrows%16==0)
#pragma unroll
        for (int r = 0; r < 8; ++r)
            pp[(size_t)r * NC] = acc[r];
    } else {
#pragma unroll
        for (int r = 0; r < 8; ++r)
            if (m0 + r + 8 * half < nrows)
                pp[(size_t)r * NC] = acc[r];
    }
}

// ---------------- edge scatter-add: G[dst] += P[src] ----------------
// One thread per (edge, 4-feature chunk); 32 consecutive threads share an edge
// for d=128 (shift=5), so edge indices are wave-uniform.
__global__ void k_aggregate(const float* __restrict__ P, const int* __restrict__ src,
                            const int* __restrict__ dst, float* __restrict__ G,
                            long long total, int shift, int d) {
    long long tid = (long long)blockIdx.x * blockDim.x + threadIdx.x;
    if (tid >= total) return;
    const int e = (int)(tid >> shift);
    const int c = (int)(tid & ((1 << shift) - 1));
    const int s = src[e];
    const int t = dst[e];
    const float4 v = *(const float4*)(P + (size_t)s * d + 4 * c);
    float* gp = G + (size_t)t * d + 4 * c;
    atomicAdd(gp + 0, v.x);
    atomicAdd(gp + 1, v.y);
    atomicAdd(gp + 2, v.z);
    atomicAdd(gp + 3, v.w);
}

// ---------------- finalize: H = (relu?)(G * nd[row] + b[col]) ----------------
__global__ void k_finalize(const float* __restrict__ G, const float* __restrict__ nd,
                           const float* __restrict__ b, float* __restrict__ H,
                           long long total, int dshift, int relu) {
    long long idx = (long long)blockIdx.x * blockDim.x + threadIdx.x;
    if (idx >= total) return;
    const int i = (int)(idx >> dshift);
    const int j = (int)(idx & ((1 << dshift) - 1));
    float v = G[idx] * nd[i] + b[j];
    if (relu) v = fmaxf(v, 0.0f);
    H[idx] = v;
}

// ---------------- launch ----------------

static inline int cdiv_ll(long long a, int b) { return (int)((a + b - 1) / b); }

extern "C" void kernel_launch(void* const* d_in, const int* in_sizes, int n_in,
                              void* d_out, int out_size, void* d_ws, size_t ws_size,
                              hipStream_t stream) {
    const float* feat = (const float*)d_in[0];
    const float* W1   = (const float*)d_in[1];
    const float* b1   = (const float*)d_in[2];
    const float* W2   = (const float*)d_in[3];
    const float* b2   = (const float*)d_in[4];
    const float* W3   = (const float*)d_in[5];
    const float* b3   = (const float*)d_in[6];
    const int*   src  = (const int*)d_in[7];
    const int*   dst  = (const int*)d_in[8];

    const int D_IN = 256, D_H = 128, D_OUT = 64;
    const int n = in_sizes[0] / D_IN;      // 100000
    const int E = in_sizes[7];             // 1600000

    // workspace layout
    float* ns   = (float*)d_ws;                    // [n]  out-degree norm
    float* nd   = ns + n;                          // [n]  in-degree norm
    float* buf0 = nd + n;                          // [n * 128]
    float* buf1 = buf0 + (size_t)n * D_H;          // [n * 128]

    float* out = (float*)d_out;

    const int mtiles = (n + 15) / 16;              // 6250

    // 1) degrees + norms (ns,nd contiguous: zero & rsqrt together)
    k_zero<<<cdiv_ll(2LL * n, 256), 256, 0, stream>>>(ns, 2LL * n);
    k_degrees<<<(E + 255) / 256, 256, 0, stream>>>(src, dst, ns, nd, E);
    k_rsqrt_norm<<<(2 * n + 255) / 256, 256, 0, stream>>>(ns, 2 * n);

    // ---- layer 1: feat[n,256] -> buf0 = (ns*feat)@W1 ; agg -> buf1 ; h1 -> buf0
    k_gemm_scaled<256, 128><<<dim3(mtiles, D_H / 16), 32, 0, stream>>>(feat, ns, W1, buf0, n);
    k_zero<<<cdiv_ll((long long)n * D_H, 256), 256, 0, stream>>>(buf1, (long long)n * D_H);
    {
        long long total = (long long)E * (D_H / 4);  // shift=5
        k_aggregate<<<cdiv_ll(total, 256), 256, 0, stream>>>(buf0, src, dst, buf1, total, 5, D_H);
    }
    k_finalize<<<cdiv_ll((long long)n * D_H, 256), 256, 0, stream>>>(buf1, nd, b1, buf0,
                                                                     (long long)n * D_H, 7, 1);

    // ---- layer 2: buf0(h1) -> buf1 = (ns*h1)@W2 ; agg -> buf0 ; h2 -> buf1
    k_gemm_scaled<128, 128><<<dim3(mtiles, D_H / 16), 32, 0, stream>>>(buf0, ns, W2, buf1, n);
    k_zero<<<cdiv_ll((long long)n * D_H, 256), 256, 0, stream>>>(buf0, (long long)n * D_H);
    {
        long long total = (long long)E * (D_H / 4);
        k_aggregate<<<cdiv_ll(total, 256), 256, 0, stream>>>(buf1, src, dst, buf0, total, 5, D_H);
    }
    k_finalize<<<cdiv_ll((long long)n * D_H, 256), 256, 0, stream>>>(buf0, nd, b2, buf1,
                                                                     (long long)n * D_H, 7, 1);

    // ---- layer 3: buf1(h2) -> buf0 = (ns*h2)@W3 [n,64] ; agg -> buf1 ; out
    k_gemm_scaled<128, 64><<<dim3(mtiles, D_OUT / 16), 32, 0, stream>>>(buf1, ns, W3, buf0, n);
    k_zero<<<cdiv_ll((long long)n * D_OUT, 256), 256, 0, stream>>>(buf1, (long long)n * D_OUT);
    {
        long long total = (long long)E * (D_OUT / 4);  // shift=4
        k_aggregate<<<cdiv_ll(total, 256), 256, 0, stream>>>(buf0, src, dst, buf1, total, 4, D_OUT);
    }
    k_finalize<<<cdiv_ll((long long)n * D_OUT, 256), 256, 0, stream>>>(buf1, nd, b3, out,
                                                                       (long long)n * D_OUT, 6, 0);
}